// Encoder_Postnet_12756052869164
// MI455X (gfx1250) — compile-verified
//
#include <hip/hip_runtime.h>

// Problem constants (from reference: B=32, P=256, REP=16, E=256, T=4096)
#define BB   32
#define PP   256
#define EE   256
#define TT   4096
#define MENC (BB * PP)          // 8192 encoder rows in combined GEMM A
#define MTOT (MENC + TT)        // 12288 total A rows (enc ++ pe)

typedef __attribute__((ext_vector_type(2))) float v2f;
typedef __attribute__((ext_vector_type(4))) float v4f;
typedef __attribute__((ext_vector_type(8))) float v8f;

// ---------------------------------------------------------------------------
// Kernel 1: per-batch sequential alignment scan.
//   idx[b,0]=0; for t>=1: ind = (align[t]==text[ind]) ? ind : min(ind+1, P-1)
// One lane per batch; text rows cached in LDS with 257 padding so lanes with
// equal `ind` hit distinct banks. `cur` caches text[ind] so the dependent LDS
// load only occurs when ind advances (runs of 16 in this data => rare).
// ---------------------------------------------------------------------------
__global__ void __launch_bounds__(32)
k_align_scan(const int* __restrict__ align, const int* __restrict__ text,
             int* __restrict__ idx)
{
    __shared__ int tx[BB * 257];
    const int lane = threadIdx.x;                 // 0..31 == batch id
    for (int i = lane; i < BB * PP; i += 32)
        tx[(i / PP) * 257 + (i % PP)] = text[i];
    __syncthreads();

    const int* al   = align + lane * TT;
    int*       outp = idx   + lane * TT;
    const int* mytx = tx + lane * 257;

    int ind = 0;
    int cur = mytx[0];
    outp[0] = 0;
    for (int t = 1; t < TT; ++t) {
        int a = al[t];
        if (a != cur) {
            ind = min(ind + 1, PP - 1);
            cur = mytx[ind];
        }
        outp[t] = ind;
    }
}

// ---------------------------------------------------------------------------
// Kernel 2: sinusoidal positional-encoding table pe[T][E].
//   pe[t, 2c]   = sin(t * exp(2c * (-ln(10000)/E)))
//   pe[t, 2c+1] = cos(  same angle  )
// ---------------------------------------------------------------------------
__global__ void __launch_bounds__(256)
k_make_pe(float* __restrict__ pe)
{
    int g = blockIdx.x * blockDim.x + threadIdx.x;   // pair index
    if (g >= TT * (EE / 2)) return;
    int t = g / (EE / 2);
    int c = g % (EE / 2);
    float div = expf((float)(2 * c) * (-9.210340371976184f / (float)EE));
    float ang = (float)t * div;
    pe[t * EE + 2 * c]     = sinf(ang);
    pe[t * EE + 2 * c + 1] = cosf(ang);
}

// ---------------------------------------------------------------------------
// Kernel 3: proj = A @ W^T  via V_WMMA_F32_16X16X4_F32 (full f32 precision).
//   A rows [0, 8192)  : encoder_out flattened (B*P, E)
//   A rows [8192,12288): pe (T, E)
//   proj[m, f] = sum_e A[m,e] * W[f,e]
// One wave per 16x16 output tile; K=256 -> 64 WMMA steps of K=4.
// ISA 32-bit 16x4 A layout: lane holds {K=2*(lane/16), +1} of row M=lane%16;
// B (4x16) mirrors with N=lane%16. C/D: acc[v] = C[v + 8*(lane/16)][lane%16].
// Grid is exact (1536 blocks * 8 waves = 12288 tiles): EXEC all-ones at WMMA.
// ---------------------------------------------------------------------------
__global__ void __launch_bounds__(256)
k_proj_gemm(const float* __restrict__ enc, const float* __restrict__ pe,
            const float* __restrict__ W, float* __restrict__ proj)
{
    const int wid  = blockIdx.x * (blockDim.x >> 5) + (threadIdx.x >> 5);
    const int lane = threadIdx.x & 31;
    const int nT   = wid & 15;        // 16 N-tiles  (N = 256)
    const int mT   = wid >> 4;        // 768 M-tiles (M = 12288)
    const int m0   = mT << 4;
    const int n0   = nT << 4;
    const int lr   = lane & 15;
    const int half = lane >> 4;
    const int koff = 2 * half;

    const float* Arow = (m0 < MENC) ? (enc + (size_t)(m0 + lr) * EE)
                                    : (pe  + (size_t)(m0 - MENC + lr) * EE);
    const float* Brow = W + (size_t)(n0 + lr) * EE;     // W row f = n0+lr

    v8f acc = {0.f, 0.f, 0.f, 0.f, 0.f, 0.f, 0.f, 0.f};
#pragma unroll 8
    for (int k = 0; k < EE; k += 4) {
        v2f a = *(const v2f*)(Arow + k + koff);
        v2f b = *(const v2f*)(Brow + k + koff);
        // 8 args: (neg_a, A, neg_b, B, c_mod, C, reuse_a, reuse_b)
        acc = __builtin_amdgcn_wmma_f32_16x16x4_f32(false, a, false, b,
                                                    (short)0, acc, false, false);
    }

    float* Crow = proj + (size_t)(m0 + 8 * half) * EE + n0 + lr;
#pragma unroll
    for (int v = 0; v < 8; ++v)
        Crow[(size_t)v * EE] = acc[v];
}

// ---------------------------------------------------------------------------
// Kernel 4: fused gather + elementwise assembly (memory-bound, 134 MB write).
// out[b,t,f] = enc[b,idx,f] + enc_proj[b,idx,f] + pe_proj[t,f]
//            + pitch[b,t]*w_pitch[f] + b_pitch[f] + beats_emb[beats[b,t],f]
//            + b_pos[f]
// 4 rows per 256-thread block; float4 per thread (64 threads cover E=256).
// ---------------------------------------------------------------------------
__global__ void __launch_bounds__(256)
k_final(const float* __restrict__ enc, const float* __restrict__ proj,
        const int* __restrict__ idx, const float* __restrict__ pitch,
        const int* __restrict__ beats,
        const float* __restrict__ wp,   const float* __restrict__ bpch,
        const float* __restrict__ bpos, const float* __restrict__ embb,
        float* __restrict__ out)
{
    const int tid = threadIdx.x;
    const int row = blockIdx.x * 4 + (tid >> 6);   // row in [0, B*T)
    const int e4  = (tid & 63) << 2;
    const int b   = row >> 12;                     // / 4096
    const int t   = row & (TT - 1);

    const int   id = idx[row];
    const float p  = pitch[row];
    const int   bt = beats[row];

    const int c = e4 >> 2;
    v4f v = ((const v4f*)(enc  + (size_t)(b * PP + id) * EE))[c]
          + ((const v4f*)(proj + (size_t)(b * PP + id) * EE))[c]
          + ((const v4f*)(proj + (size_t)(MENC + t)    * EE))[c];
    v4f w   = *(const v4f*)(wp   + e4);
    v4f add = *(const v4f*)(bpch + e4)
            + *(const v4f*)(bpos + e4)
            + *(const v4f*)(embb + bt * EE + e4);
    v = v + w * p + add;
    *(v4f*)(out + (size_t)row * EE + e4) = v;
}

// ---------------------------------------------------------------------------
extern "C" void kernel_launch(void* const* d_in, const int* in_sizes, int n_in,
                              void* d_out, int out_size, void* d_ws, size_t ws_size,
                              hipStream_t stream)
{
    const float* enc   = (const float*)d_in[0];   // (B,P,E) f32
    const int*   align = (const int*)  d_in[1];   // (B,T)   i32
    const int*   text  = (const int*)  d_in[2];   // (B,P)   i32
    const float* pitch = (const float*)d_in[3];   // (B,T,1) f32
    const int*   beats = (const int*)  d_in[4];   // (B,T,1) i32
    const float* wp    = (const float*)d_in[5];   // (E,1)   f32
    const float* bpch  = (const float*)d_in[6];   // (E,)    f32
    const float* W     = (const float*)d_in[7];   // (E,E)   f32 (f,e)
    const float* bpos  = (const float*)d_in[8];   // (E,)    f32
    const float* embb  = (const float*)d_in[9];   // (2,E)   f32
    float* out = (float*)d_out;

    // Workspace layout (all 16B-aligned): idx | pe | proj  (~17.3 MB total)
    char*  ws      = (char*)d_ws;
    int*   ws_idx  = (int*)ws;                                       // 512 KB
    float* ws_pe   = (float*)(ws + (size_t)(512 << 10));             // 4 MB
    float* ws_proj = (float*)(ws + (size_t)(512 << 10)
                                 + (size_t)TT * EE * sizeof(float)); // 12.6 MB

    k_align_scan<<<1, 32, 0, stream>>>(align, text, ws_idx);
    k_make_pe  <<<(TT * (EE / 2) + 255) / 256, 256, 0, stream>>>(ws_pe);
    k_proj_gemm<<<(MTOT / 16) * (EE / 16) / 8, 256, 0, stream>>>(enc, ws_pe, W, ws_proj);
    k_final    <<<BB * TT / 4, 256, 0, stream>>>(enc, ws_proj, ws_idx, pitch, beats,
                                                 wp, bpch, bpos, embb, out);
}